// HeteroGNN_5377299054691
// MI455X (gfx1250) — compile-verified
//
#include <hip/hip_runtime.h>

// HeteroGNN fused pipeline for MI455X (gfx1250, wave32).
//  Pass 1: zero accumulators.
//  Pass 2: edge aggregation (segment-sum of x_loc and relu(x_loc) + counts) via f32 atomics.
//  Pass 3: fully fused 3-stage per-expert-row MLP using V_WMMA_F32_16X16X4_F32,
//          all weights resident in LDS (column-major, bank-padded), per-wave LDS
//          scratch for the 16x128 intermediate tiles.

namespace {
constexpr int kNExp = 100000;
constexpr int kE    = 1000000;
constexpr int kD    = 64;    // input feature dim
constexpr int kH    = 128;   // hidden / out dim

constexpr int PAD64  = 68;   // padded K-stride for K=64 weights in LDS (bank-conflict-free)
constexpr int PAD128 = 132;  // padded K-stride for K=128 weights / scratch rows
constexpr int WAVES  = 8;    // waves per block in fused_mlp
constexpr int SCR_PER_WAVE = 16 * PAD128;

// LDS float offsets
constexpr int OFF_W0N = 0;
constexpr int OFF_W0S = OFF_W0N + kH * PAD64;    //  8704
constexpr int OFF_W1N = OFF_W0S + kH * PAD64;    // 17408
constexpr int OFF_W1S = OFF_W1N + kH * PAD64;    // 26112
constexpr int OFF_WL  = OFF_W1S + kH * PAD128;   // 43008
constexpr int OFF_SCR = OFF_WL  + kH * PAD128;   // 59904
constexpr int LDS_FLOATS = OFF_SCR + WAVES * SCR_PER_WAVE;   // 76800 floats = 307200 B
}

typedef __attribute__((ext_vector_type(2))) float v2f;
typedef __attribute__((ext_vector_type(8))) float v8f;

__device__ __forceinline__ v8f wmma4(v2f a, v2f b, v8f c) {
  // D = A(16x4) * B(4x16) + C(16x16), full f32.
  return __builtin_amdgcn_wmma_f32_16x16x4_f32(false, a, false, b, (short)0, c,
                                               false, false);
}

// ---------------------------------------------------------------- zero pass
__global__ void zero_ws_kernel(float* __restrict__ p, long n) {
  long i = (long)blockIdx.x * blockDim.x + threadIdx.x;
  long stride = (long)gridDim.x * blockDim.x;
  for (; i < n; i += stride) p[i] = 0.0f;
}

// ------------------------------------------------------------- aggregation
// 16 lanes per edge, each lane owns a float4 slice of the 64-wide feature row.
__global__ void aggregate_kernel(const float* __restrict__ x_loc,
                                 const int* __restrict__ src,
                                 const int* __restrict__ dst,
                                 float* __restrict__ acc0,
                                 float* __restrict__ acc1,
                                 float* __restrict__ cnt) {
  const long total  = (long)kE * 16;
  const long stride = (long)gridDim.x * blockDim.x;
  for (long t = (long)blockIdx.x * blockDim.x + threadIdx.x; t < total; t += stride) {
    const int e   = (int)(t >> 4);
    const int sub = (int)(t & 15);
    const int s = src[e];
    const int d = dst[e];
    const float4 v =
        reinterpret_cast<const float4*>(x_loc + (long)s * kD)[sub];
    float* p0 = acc0 + (long)d * kD + sub * 4;
    atomicAdd(p0 + 0, v.x);
    atomicAdd(p0 + 1, v.y);
    atomicAdd(p0 + 2, v.z);
    atomicAdd(p0 + 3, v.w);
    float* p1 = acc1 + (long)d * kD + sub * 4;
    atomicAdd(p1 + 0, fmaxf(v.x, 0.0f));
    atomicAdd(p1 + 1, fmaxf(v.y, 0.0f));
    atomicAdd(p1 + 2, fmaxf(v.z, 0.0f));
    atomicAdd(p1 + 3, fmaxf(v.w, 0.0f));
    if (sub == 0) atomicAdd(cnt + d, 1.0f);
  }
}

// ----------------------------------------------------------- fused 3-stage
// One wave per 16-row expert tile. WMMA f32 lane mapping (per CDNA5 ISA):
//   A 16x4:  lane L holds A[L&15][kb], A[L&15][kb+1], kb = 4*step + 2*(L>=16)
//   B 4x16:  lane L holds B[kb][L&15], B[kb+1][L&15]      (mirrored)
//   C 16x16: v8f, elem v = C[v + 8*(L>=16)][L&15]
__global__ void __launch_bounds__(256) fused_mlp_kernel(
    const float* __restrict__ acc0, const float* __restrict__ acc1,
    const float* __restrict__ cnt,  const float* __restrict__ x_exp,
    const float* __restrict__ W0n,  const float* __restrict__ W0s,
    const float* __restrict__ b0,   const float* __restrict__ W1n,
    const float* __restrict__ W1s,  const float* __restrict__ b1,
    const float* __restrict__ Wl,   const float* __restrict__ bl,
    float* __restrict__ out) {
  extern __shared__ float smem[];
  float* sW0n = smem + OFF_W0N;
  float* sW0s = smem + OFF_W0S;
  float* sW1n = smem + OFF_W1N;
  float* sW1s = smem + OFF_W1S;
  float* sWl  = smem + OFF_WL;

  const int tid = threadIdx.x;
  // Cooperative transpose-load: global row-major [K][128] -> LDS col-major
  // [128][K] with padded stride so B-fragment ds_load_b64 is conflict-free.
  for (int i = tid; i < kD * kH; i += 256) {
    const int k = i >> 7, c = i & 127;
    sW0n[c * PAD64 + k] = W0n[i];
  }
  for (int i = tid; i < kD * kH; i += 256) {
    const int k = i >> 7, c = i & 127;
    sW0s[c * PAD64 + k] = W0s[i];
  }
  for (int i = tid; i < kD * kH; i += 256) {
    const int k = i >> 7, c = i & 127;
    sW1n[c * PAD64 + k] = W1n[i];
  }
  for (int i = tid; i < kH * kH; i += 256) {
    const int k = i >> 7, c = i & 127;
    sW1s[c * PAD128 + k] = W1s[i];
  }
  for (int i = tid; i < kH * kH; i += 256) {
    const int k = i >> 7, c = i & 127;
    sWl[c * PAD128 + k] = Wl[i];
  }
  __syncthreads();

  const int wid   = tid >> 5;
  const int lane  = tid & 31;
  const int lr    = lane & 15;      // A-row within tile / B,C column within n-tile
  const int hi    = lane >> 4;      // 0 or 1
  const int khOff = hi;             // v2f-index offset inside a K=4 step
  float* sScr = smem + OFF_SCR + wid * SCR_PER_WAVE;   // [16][PAD128] x_e scratch

  const int nTiles = kNExp / 16;    // 6250
  for (int t = blockIdx.x * WAVES + wid; t < nTiles; t += gridDim.x * WAVES) {
    const int row = t * 16 + lr;
    const float inv = 1.0f / fmaxf(cnt[row], 1.0f);

    // ---------------- stage 0: x_e0 = relu(mean0@W0n + x_exp@W0s + b0) ----
    v2f aM[16], aX[16];
    {
      const v2f* pM = (const v2f*)(acc0 + (long)row * kD);
      const v2f* pX = (const v2f*)(x_exp + (long)row * kD);
#pragma unroll
      for (int kk = 0; kk < 16; ++kk) {
        aM[kk] = pM[kk * 2 + khOff] * inv;   // mean = sum / max(cnt,1)
        aX[kk] = pX[kk * 2 + khOff];
      }
    }
    for (int n = 0; n < 8; ++n) {
      const int col = n * 16 + lr;
      const float bias = b0[col];
      v8f c;
#pragma unroll
      for (int v = 0; v < 8; ++v) c[v] = bias;
      const float* wn = sW0n + col * PAD64;
      const float* ws = sW0s + col * PAD64;
#pragma unroll
      for (int kk = 0; kk < 16; ++kk) {
        const int kb = kk * 4 + khOff * 2;
        c = wmma4(aM[kk], *(const v2f*)(wn + kb), c);
        c = wmma4(aX[kk], *(const v2f*)(ws + kb), c);
      }
#pragma unroll
      for (int v = 0; v < 8; ++v)
        sScr[(v + hi * 8) * PAD128 + col] = fmaxf(c[v], 0.0f);
    }

    // ---------------- stage 1: x_e1 = relu(mean1@W1n + x_e0@W1s + b1) -----
    v2f aM1[16], aS[32];
    {
      const v2f* pM = (const v2f*)(acc1 + (long)row * kD);
#pragma unroll
      for (int kk = 0; kk < 16; ++kk) aM1[kk] = pM[kk * 2 + khOff] * inv;
      const float* pr = sScr + lr * PAD128;   // hoist full K=128 A-panel; after
#pragma unroll                                 // this, scratch may be rewritten
      for (int kk = 0; kk < 32; ++kk)
        aS[kk] = *(const v2f*)(pr + kk * 4 + khOff * 2);
    }
    for (int n = 0; n < 8; ++n) {
      const int col = n * 16 + lr;
      const float bias = b1[col];
      v8f c;
#pragma unroll
      for (int v = 0; v < 8; ++v) c[v] = bias;
      const float* wn = sW1n + col * PAD64;
#pragma unroll
      for (int kk = 0; kk < 16; ++kk)
        c = wmma4(aM1[kk], *(const v2f*)(wn + kk * 4 + khOff * 2), c);
      const float* wsl = sW1s + col * PAD128;
#pragma unroll
      for (int kk = 0; kk < 32; ++kk)
        c = wmma4(aS[kk], *(const v2f*)(wsl + kk * 4 + khOff * 2), c);
#pragma unroll
      for (int v = 0; v < 8; ++v)
        sScr[(v + hi * 8) * PAD128 + col] = fmaxf(c[v], 0.0f);
    }

    // ---------------- stage 2: out = x_e1 @ W_lin + b_lin -----------------
    v2f aE[32];
    {
      const float* pr = sScr + lr * PAD128;
#pragma unroll
      for (int kk = 0; kk < 32; ++kk)
        aE[kk] = *(const v2f*)(pr + kk * 4 + khOff * 2);
    }
    for (int n = 0; n < 8; ++n) {
      const int col = n * 16 + lr;
      const float bias = bl[col];
      v8f c;
#pragma unroll
      for (int v = 0; v < 8; ++v) c[v] = bias;
      const float* w = sWl + col * PAD128;
#pragma unroll
      for (int kk = 0; kk < 32; ++kk)
        c = wmma4(aE[kk], *(const v2f*)(w + kk * 4 + khOff * 2), c);
#pragma unroll
      for (int v = 0; v < 8; ++v)
        out[(long)(t * 16 + v + hi * 8) * kH + col] = c[v];
    }
  }
}

// ---------------------------------------------------------------------------
extern "C" void kernel_launch(void* const* d_in, const int* in_sizes, int n_in,
                              void* d_out, int out_size, void* d_ws,
                              size_t ws_size, hipStream_t stream) {
  const float* x_loc = (const float*)d_in[0];
  const float* x_exp = (const float*)d_in[1];
  const float* W0n   = (const float*)d_in[2];
  const float* W0s   = (const float*)d_in[3];
  const float* b0    = (const float*)d_in[4];
  const float* W1n   = (const float*)d_in[5];
  const float* W1s   = (const float*)d_in[6];
  const float* b1    = (const float*)d_in[7];
  const float* Wl    = (const float*)d_in[8];
  const float* bl    = (const float*)d_in[9];
  const int* edges   = (const int*)d_in[10];
  const int* src = edges;            // edge_index[0]
  const int* dst = edges + kE;       // edge_index[1]

  float* acc0 = (float*)d_ws;                       // [N_EXP][64] sum x_loc
  float* acc1 = acc0 + (long)kNExp * kD;            // [N_EXP][64] sum relu(x_loc)
  float* cnt  = acc1 + (long)kNExp * kD;            // [N_EXP]
  const long nzero = (long)kNExp * kD * 2 + kNExp;

  zero_ws_kernel<<<2048, 256, 0, stream>>>(acc0, nzero);
  aggregate_kernel<<<4096, 256, 0, stream>>>(x_loc, src, dst, acc0, acc1, cnt);

  const size_t ldsBytes = (size_t)LDS_FLOATS * sizeof(float);   // 307200 B
  const int nTiles = kNExp / 16;
  const int grid = (nTiles + WAVES - 1) / WAVES;                // 782 blocks
  fused_mlp_kernel<<<grid, 256, ldsBytes, stream>>>(
      acc0, acc1, cnt, x_exp, W0n, W0s, b0, W1n, W1s, b1, Wl, bl,
      (float*)d_out);
}